// GcnEdgeConvNet_31593779430169
// MI455X (gfx1250) — compile-verified
//
#include <hip/hip_runtime.h>

// ---------------------------------------------------------------------------
// GcnEdgeConvNet forward for MI455X (gfx1250), FP32 end-to-end.
// All dense GEMMs (node convs + edge linears) run on V_WMMA_F32_16X16X4_F32.
// Guard-free inner loops: row tiles are exact (N%16==0, E%16==0), K-dim is
// zero-padded in both activations (padded strides, memset) and weights
// (repacked Kp x Mp panels), so every load is unconditional.
// ---------------------------------------------------------------------------

typedef float v2f __attribute__((ext_vector_type(2)));
typedef float v8f __attribute__((ext_vector_type(8)));

#define N_NODES 50000
#define N_EDGES 800000
#define HS 48   // node-feature row stride (max F=45, padded, %4==0)
#define ES 32   // edge-feature row stride (max F=25, padded, %4==0)

constexpr int round4(int x)  { return (x + 3)  & ~3;  }
constexpr int round16(int x) { return (x + 15) & ~15; }

// ---------------- degree / normalization -----------------------------------
__global__ void k_deg_init(float* __restrict__ deg) {
  int i = blockIdx.x * blockDim.x + threadIdx.x;
  if (i < N_NODES) deg[i] = 1.0f;  // self loop
}
__global__ void k_deg_count(const int* __restrict__ col, float* __restrict__ deg) {
  int e = blockIdx.x * blockDim.x + threadIdx.x;
  if (e < N_EDGES) atomicAdd(&deg[col[e]], 1.0f);
}
__global__ void k_dinv(float* __restrict__ deg) {
  int i = blockIdx.x * blockDim.x + threadIdx.x;
  if (i < N_NODES) deg[i] = rsqrtf(deg[i]);  // deg >= 1 always
}

// ---------------- weight repack into zero-padded Kp x Mp panel --------------
__global__ void k_padW(const float* __restrict__ W, int K, int M,
                       float* __restrict__ Wp, int Mp, int total) {
  int idx = blockIdx.x * blockDim.x + threadIdx.x;
  if (idx >= total) return;
  int k = idx / Mp, n = idx - k * Mp;
  Wp[idx] = (k < K && n < M) ? W[k * M + n] : 0.0f;
}

// ---------------- copy raw edge features into padded buffer -----------------
__global__ void k_copy_e(const float* __restrict__ e0, float* __restrict__ eo) {
  int idx = blockIdx.x * blockDim.x + threadIdx.x;
  if (idx >= N_EDGES * 10) return;
  int e = idx / 10, f = idx - e * 10;
  eo[(size_t)e * ES + f] = e0[idx];
}

// ---------------- dense GEMM: Y[N,M] = X[N,Kp] @ Wp[Kp,Mp] (WMMA f32) -------
// One wave32 per 16x16 output tile; no bounds checks inside the K loop.
template <int Kp, int Mp, int M>
__global__ __launch_bounds__(32) void k_gemm16(
    const float* __restrict__ X, int ldx,
    const float* __restrict__ Wp,
    float* __restrict__ Y) {
  const int lane = threadIdx.x;
  const int half = lane >> 4;   // half 0: K+0/K+1 ; half 1: K+2/K+3
  const int l    = lane & 15;
  const int arow = blockIdx.x * 16 + l;   // always < N (N%16==0)
  const int bcol = blockIdx.y * 16 + l;   // always < Mp

  const float* xp = X + (size_t)arow * ldx + half * 2;
  const float* wp = Wp + (size_t)(half * 2) * Mp + bcol;

  v8f c = {};
  for (int k0 = 0; k0 < Kp; k0 += 4) {
    v2f a = *(const v2f*)(xp + k0);       // contiguous pair -> b64 load
    v2f b;
    b.x = wp[(size_t)k0 * Mp];
    b.y = wp[(size_t)k0 * Mp + Mp];
    c = __builtin_amdgcn_wmma_f32_16x16x4_f32(
        false, a, false, b, (short)0, c, false, false);
  }
  const int n = blockIdx.y * 16 + l;
  if (n < M) {
#pragma unroll
    for (int v = 0; v < 8; v++)
      Y[(size_t)(blockIdx.x * 16 + v + half * 8) * HS + n] = c[v];
  }
}

// ---------------- GCN aggregation -------------------------------------------
__global__ void k_selfloop(const float* __restrict__ h, const float* __restrict__ dinv,
                           float* __restrict__ out, int M) {
  int idx = blockIdx.x * blockDim.x + threadIdx.x;
  if (idx >= N_NODES * M) return;
  int i = idx / M, f = idx - i * M;
  float d = dinv[i];
  out[(size_t)i * HS + f] = h[(size_t)i * HS + f] * d * d;
}

__global__ void k_scatter(const float* __restrict__ h,
                          const int* __restrict__ row, const int* __restrict__ col,
                          const float* __restrict__ dinv,
                          float* __restrict__ out, int M) {
  int e = blockIdx.x * blockDim.x + threadIdx.x;
  if (e >= N_EDGES) return;
  int r = row[e], c = col[e];
  float nrm = dinv[r] * dinv[c];
  const float* hr = h + (size_t)r * HS;
  float* oc = out + (size_t)c * HS;
  for (int f = 0; f < M; f++) atomicAdd(&oc[f], hr[f] * nrm);
}

__global__ void k_bias_relu(float* __restrict__ y, const float* __restrict__ b, int M) {
  int idx = blockIdx.x * blockDim.x + threadIdx.x;
  if (idx >= N_NODES * M) return;
  int i = idx / M, f = idx - i * M;
  float v = y[(size_t)i * HS + f] + b[f];
  y[(size_t)i * HS + f] = v > 0.0f ? v : 0.0f;
}

// ---------------- edge linear: relu([h_i || h_j || e] @ W + b) (WMMA f32) ---
// Branch-free region select on base pointers; all loads unconditional.
template <int Fh, int Fe, int M>
__global__ __launch_bounds__(32) void k_edgelin16(
    const float* __restrict__ H,
    const float* __restrict__ Ein,
    const int* __restrict__ row, const int* __restrict__ col,
    const float* __restrict__ Wp, const float* __restrict__ bias,
    float* __restrict__ Eout) {
  constexpr int K  = 2 * Fh + Fe;
  constexpr int Kp = round4(K);
  constexpr int Mp = round16(M);
  static_assert(Kp - 2 * Fh <= ES, "edge pad read must stay inside padded row");

  const int lane = threadIdx.x;
  const int half = lane >> 4;
  const int l    = lane & 15;

  const int ei = blockIdx.x * 16 + l;   // always < E (E%16==0)
  const int ri = row[ei], ci = col[ei];
  const float* baseA = H + (size_t)ri * HS;              // k in [0, Fh)
  const float* baseB = H + (size_t)ci * HS - Fh;         // k in [Fh, 2Fh)
  const float* baseE = Ein + (size_t)ei * ES - 2 * Fh;   // k in [2Fh, Kp)

  const int bcol = blockIdx.y * 16 + l;
  const float* wp = Wp + (size_t)(half * 2) * Mp + bcol;

  v8f c = {};
  for (int k0 = 0; k0 < Kp; k0 += 4) {
    const int ka = k0 + half * 2;
    const int kb = ka + 1;
    const float* pa = (ka < Fh) ? (baseA + ka)
                                : ((ka < 2 * Fh) ? (baseB + ka) : (baseE + ka));
    const float* pb = (kb < Fh) ? (baseA + kb)
                                : ((kb < 2 * Fh) ? (baseB + kb) : (baseE + kb));
    v2f a, b;
    a.x = *pa;
    a.y = *pb;
    b.x = wp[(size_t)k0 * Mp];
    b.y = wp[(size_t)k0 * Mp + Mp];
    c = __builtin_amdgcn_wmma_f32_16x16x4_f32(
        false, a, false, b, (short)0, c, false, false);
  }
  const int n = blockIdx.y * 16 + l;
  if (n < M) {
    const float bb = bias[n];
#pragma unroll
    for (int v = 0; v < 8; v++) {
      float r = c[v] + bb;
      Eout[(size_t)(blockIdx.x * 16 + v + half * 8) * ES + n] = r > 0.0f ? r : 0.0f;
    }
  }
}

// ---------------- 2-class softmax over relu'd logits ------------------------
__global__ void k_softmax2(const float* __restrict__ logits, float* __restrict__ out) {
  int e = blockIdx.x * blockDim.x + threadIdx.x;
  if (e >= N_EDGES) return;
  float a = logits[(size_t)e * ES + 0];
  float b = logits[(size_t)e * ES + 1];
  float m = fmaxf(a, b);
  float ea = __expf(a - m), eb = __expf(b - m);
  float inv = 1.0f / (ea + eb);
  out[2 * e + 0] = ea * inv;
  out[2 * e + 1] = eb * inv;
}

// ---------------- host-side layer helpers -----------------------------------
template <int K, int M>
static void gcn_layer(const float* xin, int ldx, const float* Wp, const float* b,
                      float* hT, float* xout,
                      const int* row, const int* col, const float* dinv,
                      hipStream_t stream) {
  constexpr int Kp = round4(K);
  constexpr int Mp = round16(M);
  dim3 g(N_NODES / 16, Mp / 16);
  k_gemm16<Kp, Mp, M><<<g, 32, 0, stream>>>(xin, ldx, Wp, hT);
  int tot = N_NODES * M;
  k_selfloop<<<(tot + 255) / 256, 256, 0, stream>>>(hT, dinv, xout, M);
  k_scatter<<<(N_EDGES + 255) / 256, 256, 0, stream>>>(hT, row, col, dinv, xout, M);
  k_bias_relu<<<(tot + 255) / 256, 256, 0, stream>>>(xout, b, M);
}

template <int Fh, int Fe, int M>
static void elin_layer(const float* H, const float* Ein,
                       const int* row, const int* col,
                       const float* Wp, const float* b, float* Eout,
                       hipStream_t stream) {
  constexpr int Mp = round16(M);
  dim3 g(N_EDGES / 16, Mp / 16);
  k_edgelin16<Fh, Fe, M><<<g, 32, 0, stream>>>(H, Ein, row, col, Wp, b, Eout);
}

// ---------------------------------------------------------------------------
extern "C" void kernel_launch(void* const* d_in, const int* in_sizes, int n_in,
                              void* d_out, int out_size, void* d_ws, size_t ws_size,
                              hipStream_t stream) {
  const float* x   = (const float*)d_in[0];   // [N,16]
  const float* e0  = (const float*)d_in[1];   // [E,10]
  const int*   ein = (const int*)d_in[2];     // [2,E]
  const int* row = ein;
  const int* col = ein + N_EDGES;

  const float *W1 = (const float*)d_in[3],  *b1 = (const float*)d_in[4];
  const float *W2 = (const float*)d_in[5],  *b2 = (const float*)d_in[6];
  const float *W3 = (const float*)d_in[7],  *b3 = (const float*)d_in[8];
  const float *Wn4 = (const float*)d_in[9],  *bn4 = (const float*)d_in[10];
  const float *We4 = (const float*)d_in[11], *be4 = (const float*)d_in[12];
  const float *Wn5 = (const float*)d_in[13], *bn5 = (const float*)d_in[14];
  const float *We5 = (const float*)d_in[15], *be5 = (const float*)d_in[16];
  const float *Wn6 = (const float*)d_in[17], *bn6 = (const float*)d_in[18];
  const float *We6 = (const float*)d_in[19], *be6 = (const float*)d_in[20];
  const float *We7 = (const float*)d_in[21], *be7 = (const float*)d_in[22];

  // workspace carve-up (256B aligned)
  char* base = (char*)d_ws;
  char* p = base;
  auto take = [&](size_t elems) -> float* {
    float* r = (float*)p;
    p += (elems * 4 + 255) & ~(size_t)255;
    return r;
  };
  float* dinv = take(N_NODES);
  float* hT   = take((size_t)N_NODES * HS);
  float* xA   = take((size_t)N_NODES * HS);
  float* xB   = take((size_t)N_NODES * HS);
  float* eA   = take((size_t)N_EDGES * ES);
  float* eB   = take((size_t)N_EDGES * ES);
  // padded weight panels
  auto takeW = [&](int K, int M) -> float* { return take((size_t)round4(K) * round16(M)); };
  float* W1p  = takeW(16, 15);
  float* W2p  = takeW(15, 25);
  float* W3p  = takeW(25, 30);
  float* Wn4p = takeW(30, 35);
  float* Wn5p = takeW(35, 40);
  float* Wn6p = takeW(40, 45);
  float* We4p = takeW(70, 15);
  float* We5p = takeW(85, 20);
  float* We6p = takeW(100, 25);
  float* We7p = takeW(115, 2);
  size_t used = (size_t)(p - base);

  const int TB = 256;
  // zero everything (pads must be 0; graph-capturable memset node)
  hipMemsetAsync(d_ws, 0, used, stream);

  // GCN normalization: dinv = (deg+1)^(-1/2)
  k_deg_init<<<(N_NODES + TB - 1) / TB, TB, 0, stream>>>(dinv);
  k_deg_count<<<(N_EDGES + TB - 1) / TB, TB, 0, stream>>>(col, dinv);
  k_dinv<<<(N_NODES + TB - 1) / TB, TB, 0, stream>>>(dinv);

  // weight repack
  auto padW = [&](const float* W, int K, int M, float* Wp) {
    int tot = round4(K) * round16(M);
    k_padW<<<(tot + TB - 1) / TB, TB, 0, stream>>>(W, K, M, Wp, round16(M), tot);
  };
  padW(W1, 16, 15, W1p);   padW(W2, 15, 25, W2p);   padW(W3, 25, 30, W3p);
  padW(Wn4, 30, 35, Wn4p); padW(Wn5, 35, 40, Wn5p); padW(Wn6, 40, 45, Wn6p);
  padW(We4, 70, 15, We4p); padW(We5, 85, 20, We5p);
  padW(We6, 100, 25, We6p); padW(We7, 115, 2, We7p);

  // raw edge features -> padded buffer eA
  k_copy_e<<<(N_EDGES * 10 + TB - 1) / TB, TB, 0, stream>>>(e0, eA);

  // layers 1-3: GCN node convs
  gcn_layer<16, 15>(x,  16, W1p, b1, hT, xA, row, col, dinv, stream);
  gcn_layer<15, 25>(xA, HS, W2p, b2, hT, xB, row, col, dinv, stream);
  gcn_layer<25, 30>(xB, HS, W3p, b3, hT, xA, row, col, dinv, stream);
  // layer 4: node 30->35 ; edge [30||30||10]->15 (both use pre-update xA)
  gcn_layer<30, 35>(xA, HS, Wn4p, bn4, hT, xB, row, col, dinv, stream);
  elin_layer<30, 10, 15>(xA, eA, row, col, We4p, be4, eB, stream);
  // layer 5: node 35->40 ; edge [35||35||15]->20
  gcn_layer<35, 40>(xB, HS, Wn5p, bn5, hT, xA, row, col, dinv, stream);
  elin_layer<35, 15, 20>(xB, eB, row, col, We5p, be5, eA, stream);
  // layer 6: node 40->45 ; edge [40||40||20]->25
  gcn_layer<40, 45>(xA, HS, Wn6p, bn6, hT, xB, row, col, dinv, stream);
  elin_layer<40, 20, 25>(xA, eA, row, col, We6p, be6, eB, stream);
  // layer 7: edge classifier [45||45||25]->2, then softmax
  elin_layer<45, 25, 2>(xB, eB, row, col, We7p, be7, eA, stream);
  k_softmax2<<<(N_EDGES + TB - 1) / TB, TB, 0, stream>>>(eA, (float*)d_out);
}